// TriPlaneEmbedding_63934883168881
// MI455X (gfx1250) — compile-verified
//
#include <hip/hip_runtime.h>

namespace {
constexpr int CH = 32;                 // channels per plane
constexpr int R  = 257;                // plane is (RES+1) x (RES+1) = 257x257
constexpr int HW = R * R;              // 66049 texels
constexpr long long PLANE_ELTS = (long long)HW * CH;
}

// ---------------------------------------------------------------------------
// Pass 1: transpose planes [C, H, W] -> [H*W, C] so one texel's 32 channels
// are one contiguous 128-byte cacheline (one coalesced load per wave32).
// LDS-tiled: coalesced reads (64 consecutive texels per channel) and
// coalesced writes (32 consecutive channels per wave). tile padded to 33
// floats/row -> stride 33 coprime with 64 banks: conflict-free.
// ---------------------------------------------------------------------------
__global__ void tp_transpose(const float* __restrict__ s0,
                             const float* __restrict__ s1,
                             const float* __restrict__ s2,
                             float* __restrict__ d0,
                             float* __restrict__ d1,
                             float* __restrict__ d2) {
  __shared__ float tile[64][CH + 1];
  const float* __restrict__ src = (blockIdx.y == 0) ? s0 : (blockIdx.y == 1 ? s1 : s2);
  float* __restrict__ dst       = (blockIdx.y == 0) ? d0 : (blockIdx.y == 1 ? d1 : d2);

  const int tid  = threadIdx.x;       // 0..255
  const int pos0 = blockIdx.x * 64;

#pragma unroll
  for (int k = 0; k < 8; ++k) {       // read: 64 threads share a channel
    int c   = k * 4 + (tid >> 6);     // 0..31
    int pl  = tid & 63;
    int pos = pos0 + pl;
    tile[pl][c] = (pos < HW) ? src[c * HW + pos] : 0.0f;
  }
  __syncthreads();
#pragma unroll
  for (int k = 0; k < 8; ++k) {       // write: wave writes 32 contiguous ch
    int pl  = k * 8 + (tid >> 5);
    int c   = tid & 31;
    int pos = pos0 + pl;
    if (pos < HW) dst[pos * CH + c] = tile[pl][c];
  }
}

// ---------------------------------------------------------------------------
// Bilinear sample matching jax clip((c+1)*0.5*(W-1), 0, W-1) semantics
// (grid_sample align_corners=True, border padding).
// Base-plus-delta addressing: dx/dy are 0 at the clamped border, where the
// corresponding weight is 0 anyway, so the degenerate aliasing is exact.
// ---------------------------------------------------------------------------
template <bool HWC>
__device__ __forceinline__ float sample_plane(const float* __restrict__ plane,
                                              float cx, float cy, int lane) {
  float x = fminf(fmaxf((cx + 1.0f) * (0.5f * (float)(R - 1)), 0.0f), (float)(R - 1));
  float y = fminf(fmaxf((cy + 1.0f) * (0.5f * (float)(R - 1)), 0.0f), (float)(R - 1));
  float xf = floorf(x), yf = floorf(y);
  float wx = x - xf,   wy = y - yf;
  int x0 = (int)xf,    y0 = (int)yf;
  int x1 = min(x0 + 1, R - 1);
  int y1 = min(y0 + 1, R - 1);

  constexpr int SX = HWC ? CH     : 1;   // stride in x
  constexpr int SY = HWC ? R * CH : R;   // stride in y
  int base = HWC ? ((y0 * R + x0) * CH + lane)
                 : (lane * HW + y0 * R + x0);
  int dx = (x1 - x0) * SX;               // 0 or SX
  int dy = (y1 - y0) * SY;               // 0 or SY

  float f00 = plane[base];
  float f01 = plane[base + dx];
  float f10 = plane[base + dy];
  float f11 = plane[base + dy + dx];
  float top = fmaf(wx, f01 - f00, f00);
  float bot = fmaf(wx, f11 - f10, f10);
  return fmaf(wy, bot - top, top);
}

// ---------------------------------------------------------------------------
// Pass 2: one wave32 per point (lane == channel), 4 points per wave.
// Fast path (always taken for npts % 32 == 0): straight-line unrolled body so
// the scheduler can batch all 4x12 coalesced 128B gathers for deep MLP while
// the wave sits on L2 latency. Non-temporal 128B stores keep the 768MB output
// stream from evicting the 25MB of L2-resident plane data.
// ---------------------------------------------------------------------------
template <bool HWC>
__global__ void __launch_bounds__(256)
tp_sample(const float* __restrict__ xyz,
          const float* __restrict__ pxy,
          const float* __restrict__ pyz,
          const float* __restrict__ pxz,
          float* __restrict__ out, int npts) {
  const int lane = threadIdx.x & 31;
  const int wave = threadIdx.x >> 5;
  constexpr int PPW = 4;                               // points per wave
  const int p0 = (blockIdx.x * (int)(blockDim.x >> 5) + wave) * PPW;

  if (p0 + PPW <= npts) {
    // gfx1250 global_prefetch_b8: nudge this wave's coord cachelines inward
    __builtin_prefetch(&xyz[p0 * 3], 0, 3);

    float px[PPW], py[PPW], pz[PPW];
#pragma unroll
    for (int i = 0; i < PPW; ++i) {                    // streaming coord reads
      px[i] = __builtin_nontemporal_load(&xyz[(p0 + i) * 3 + 0]);
      py[i] = __builtin_nontemporal_load(&xyz[(p0 + i) * 3 + 1]);
      pz[i] = __builtin_nontemporal_load(&xyz[(p0 + i) * 3 + 2]);
    }

    float vxy[PPW], vyz[PPW], vxz[PPW];
#pragma unroll
    for (int i = 0; i < PPW; ++i) {
      vxy[i] = sample_plane<HWC>(pxy, px[i], py[i], lane);  // cx=px, cy=py
      vyz[i] = sample_plane<HWC>(pyz, py[i], pz[i], lane);  // cx=py, cy=pz
      vxz[i] = sample_plane<HWC>(pxz, px[i], pz[i], lane);  // cx=px, cy=pz
    }

#pragma unroll
    for (int i = 0; i < PPW; ++i) {
      float* o = out + (long long)(p0 + i) * (3 * CH);
      __builtin_nontemporal_store(vxy[i], &o[lane]);
      __builtin_nontemporal_store(vyz[i], &o[CH + lane]);
      __builtin_nontemporal_store(vxz[i], &o[2 * CH + lane]);
    }
  } else {
    // remainder path (never taken when npts is a multiple of 32)
    for (int i = 0; i < PPW; ++i) {
      int p = p0 + i;
      if (p >= npts) return;
      float px = __builtin_nontemporal_load(&xyz[p * 3 + 0]);
      float py = __builtin_nontemporal_load(&xyz[p * 3 + 1]);
      float pz = __builtin_nontemporal_load(&xyz[p * 3 + 2]);
      float vxy = sample_plane<HWC>(pxy, px, py, lane);
      float vyz = sample_plane<HWC>(pyz, py, pz, lane);
      float vxz = sample_plane<HWC>(pxz, px, pz, lane);
      float* o = out + (long long)p * (3 * CH);
      __builtin_nontemporal_store(vxy, &o[lane]);
      __builtin_nontemporal_store(vyz, &o[CH + lane]);
      __builtin_nontemporal_store(vxz, &o[2 * CH + lane]);
    }
  }
}

// ---------------------------------------------------------------------------
extern "C" void kernel_launch(void* const* d_in, const int* in_sizes, int n_in,
                              void* d_out, int out_size, void* d_ws, size_t ws_size,
                              hipStream_t stream) {
  (void)n_in; (void)out_size;
  const float* xyz = (const float*)d_in[0];
  const float* xy  = (const float*)d_in[1];
  const float* yz  = (const float*)d_in[2];
  const float* xz  = (const float*)d_in[3];
  float* out = (float*)d_out;

  const int npts = in_sizes[0] / 3;
  const int pts_per_block = (256 / 32) * 4;            // 8 waves * 4 points
  const int nblk = (npts + pts_per_block - 1) / pts_per_block;

  const size_t need = 3 * (size_t)PLANE_ELTS * sizeof(float);  // ~25.4 MB
  if (ws_size >= need) {
    float* t0 = (float*)d_ws;
    float* t1 = t0 + PLANE_ELTS;
    float* t2 = t1 + PLANE_ELTS;
    dim3 tgrid((HW + 63) / 64, 3);
    tp_transpose<<<tgrid, 256, 0, stream>>>(xy, yz, xz, t0, t1, t2);
    tp_sample<true><<<nblk, 256, 0, stream>>>(xyz, t0, t1, t2, out, npts);
  } else {
    // workspace too small: sample original [C,H,W] layout directly
    tp_sample<false><<<nblk, 256, 0, stream>>>(xyz, xy, yz, xz, out, npts);
  }
}